// SNN_85323820302471
// MI455X (gfx1250) — compile-verified
//
#include <hip/hip_runtime.h>

// ---------------- problem constants (from reference) ----------------
constexpr int   B   = 32;
constexpr int   N   = 4096;     // internal neurons
constexpr int   D   = 8;        // internal delay bins
constexpr int   NE  = 2048;     // external sources
constexpr int   DE  = 4;        // external delay bins
constexpr int   BN  = B * N;    // 131072
constexpr int   BNE = B * NE;   // 65536

constexpr float ALPHAc = 0.95f;
constexpr float RHOc   = 0.98f;
constexpr float BETAc  = 1.8f;
constexpr float TH0c   = 1.0f;

// K-split for occupancy: 32 column-blocks x 8 K-chunks = 256 workgroups
constexpr int KSPLIT  = 8;
constexpr int KC_INT  = N  / KSPLIT;  // 512  (multiple of 4)
constexpr int KC_EXT  = NE / KSPLIT;  // 256  (multiple of 4)

typedef __attribute__((ext_vector_type(2))) float v2f;
typedef __attribute__((ext_vector_type(8))) float v8f;

// ---------------------------------------------------------------------
// Kernel 1: ALIF spike + a_new + V_new seed + internal delay gather
//   out[0] = X, out[2] = rho*a + X, out[1] = alpha*V*(1-X)  (GEMM adds on top)
//   xint[b,i] = dmap_int[0,i]*X[b,i] + sum_{d=1..7} dmap_int[d,i]*Xd[d-1,b,i]
// ---------------------------------------------------------------------
__global__ __launch_bounds__(256) void snn_spike_kernel(
    const float* __restrict__ V, const float* __restrict__ a,
    const float* __restrict__ Xd, const float* __restrict__ dmap_int,
    float* __restrict__ out, float* __restrict__ xint)
{
    const int idx = blockIdx.x * 256 + threadIdx.x;   // idx = b*N + i, grid exact
    const int i   = idx & (N - 1);

    const float v  = V[idx];
    const float aa = a[idx];
    const float X  = (v - (TH0c + BETAc * aa) >= 0.0f) ? 1.0f : 0.0f;

    out[idx]          = X;                       // X
    out[2 * BN + idx] = RHOc * aa + X;           // a_new
    out[BN + idx]     = ALPHAc * v * (1.0f - X); // V_new seed (current added by GEMM)

    float s = dmap_int[i] * X;                   // delay bin 0 = current spikes
    #pragma unroll
    for (int d = 1; d < D; ++d)
        s += dmap_int[d * N + i] * Xd[(size_t)(d - 1) * BN + idx];
    xint[idx] = s;
}

// ---------------------------------------------------------------------
// Kernel 2: external delay gather  xext[b,i] = sum_d dmap_ext[d,i]*Xext[d,b,i]
// ---------------------------------------------------------------------
__global__ __launch_bounds__(256) void snn_ext_kernel(
    const float* __restrict__ Xext, const float* __restrict__ dmap_ext,
    float* __restrict__ xext)
{
    const int idx = blockIdx.x * 256 + threadIdx.x;   // idx = b*NE + i
    const int i   = idx & (NE - 1);
    float s = 0.0f;
    #pragma unroll
    for (int d = 0; d < DE; ++d)
        s += dmap_ext[d * NE + i] * Xext[(size_t)d * BNE + idx];
    xext[idx] = s;
}

// ---------------------------------------------------------------------
// Kernel 3: weight-streaming WMMA GEMM
//   current[b,j] = sum_i xint[b,i]*Wint[i,j] + sum_i xext[b,i]*Wext[i,j]
// Wave = one 32x16 C tile (two v8f accumulators). Block = 8 waves = 128 cols.
// grid = (N/128, KSPLIT); partials combined via global_atomic_add_f32 into out[1].
// W loaded non-temporal (read-once stream); activations stay resident in L2.
// ---------------------------------------------------------------------
__global__ __launch_bounds__(256) void snn_gemm_kernel(
    const float* __restrict__ xint, const float* __restrict__ xext,
    const float* __restrict__ Wint, const float* __restrict__ Wext,
    float* __restrict__ outV)
{
    const int lane  = threadIdx.x & 31;
    const int wave  = threadIdx.x >> 5;
    const int nlane = lane & 15;             // column-within-tile / M-within-tile
    const int khalf = (lane >> 4) << 1;      // 0 (lanes 0-15) or 2 (lanes 16-31)
    const int col   = blockIdx.x * 128 + wave * 16 + nlane;
    const int mbase = (lane >> 4) << 3;      // C/D rows: m = vgpr + mbase

    v8f c0 = {0.f,0.f,0.f,0.f,0.f,0.f,0.f,0.f};  // batch rows 0..15
    v8f c1 = {0.f,0.f,0.f,0.f,0.f,0.f,0.f,0.f};  // batch rows 16..31

    // ---- internal weights: K chunk [by*KC_INT, +KC_INT) over Wint [N x N] ----
    {
        const float* arow0 = xint + (size_t)nlane * N;         // A[m=nlane, k]
        const float* arow1 = xint + (size_t)(nlane + 16) * N;  // batch rows 16..31
        const int kbeg = blockIdx.y * KC_INT;
        #pragma unroll 4
        for (int k = kbeg; k < kbeg + KC_INT; k += 4) {
            const int ka = k + khalf;
            v2f a0 = *(const v2f*)(arow0 + ka);                // contiguous float2
            v2f a1 = *(const v2f*)(arow1 + ka);
            v2f b;
            b.x = __builtin_nontemporal_load(Wint + (size_t)ka * N + col);
            b.y = __builtin_nontemporal_load(Wint + (size_t)(ka + 1) * N + col);
            c0 = __builtin_amdgcn_wmma_f32_16x16x4_f32(false, a0, false, b,
                                                       (short)0, c0, false, false);
            c1 = __builtin_amdgcn_wmma_f32_16x16x4_f32(false, a1, false, b,
                                                       (short)0, c1, false, false);
        }
    }

    // ---- external weights: K chunk over Wext [NE x N] ----
    {
        const float* arow0 = xext + (size_t)nlane * NE;
        const float* arow1 = xext + (size_t)(nlane + 16) * NE;
        const int kbeg = blockIdx.y * KC_EXT;
        #pragma unroll 4
        for (int k = kbeg; k < kbeg + KC_EXT; k += 4) {
            const int ka = k + khalf;
            v2f a0 = *(const v2f*)(arow0 + ka);
            v2f a1 = *(const v2f*)(arow1 + ka);
            v2f b;
            b.x = __builtin_nontemporal_load(Wext + (size_t)ka * N + col);
            b.y = __builtin_nontemporal_load(Wext + (size_t)(ka + 1) * N + col);
            c0 = __builtin_amdgcn_wmma_f32_16x16x4_f32(false, a0, false, b,
                                                       (short)0, c0, false, false);
            c1 = __builtin_amdgcn_wmma_f32_16x16x4_f32(false, a1, false, b,
                                                       (short)0, c1, false, false);
        }
    }

    // ---- epilogue: K-split partial sums -> out[1] via f32 atomics ----
    union { v8f v; float f[8]; } u0, u1;
    u0.v = c0; u1.v = c1;
    #pragma unroll
    for (int r = 0; r < 8; ++r) {
        atomicAdd(outV + (size_t)(mbase + r) * N + col,      u0.f[r]);
        atomicAdd(outV + (size_t)(mbase + r + 16) * N + col, u1.f[r]);
    }
}

// ---------------------------------------------------------------------
extern "C" void kernel_launch(void* const* d_in, const int* in_sizes, int n_in,
                              void* d_out, int out_size, void* d_ws, size_t ws_size,
                              hipStream_t stream)
{
    const float* V        = (const float*)d_in[0];  // [B,N]
    const float* a        = (const float*)d_in[1];  // [B,N]
    const float* Xd       = (const float*)d_in[2];  // [D,B,N]
    const float* Xext     = (const float*)d_in[3];  // [DE,B,NE]
    const float* Wint     = (const float*)d_in[4];  // [N,N]
    const float* Wext     = (const float*)d_in[5];  // [NE,N]
    const float* dmap_int = (const float*)d_in[6];  // [D,N]
    const float* dmap_ext = (const float*)d_in[7];  // [DE,NE]
    float* out = (float*)d_out;                     // [3,B,N] = X, V_new, a_new

    float* xint = (float*)d_ws;                     // [B,N]  fp32
    float* xext = xint + BN;                        // [B,NE] fp32  (ws: 768 KB total)

    snn_spike_kernel<<<BN / 256, 256, 0, stream>>>(V, a, Xd, dmap_int, out, xint);
    snn_ext_kernel<<<BNE / 256, 256, 0, stream>>>(Xext, dmap_ext, xext);

    dim3 grid(N / 128, KSPLIT);                     // 32 x 8 = 256 workgroups
    snn_gemm_kernel<<<grid, 256, 0, stream>>>(xint, xext, Wint, Wext, out + BN);
}